// Expansion_84748294684812
// MI455X (gfx1250) — compile-verified
//
#include <hip/hip_runtime.h>

// ---------------------------------------------------------------------------
// e3nn "Expansion": out[b, 16j^2+u(2j+1)+p, 16k^2+v(2k+1)+q] =
//   sum_{(i,j,k) in INSTR} sum_{w,r} W_n[w,u,v] * w3j_n[p,q,r] * x_i[b,w,r] / 128
//
// Plan (MI455X, gfx1250):
//  kernel 0: e3nn real Wigner-3j tensors on-device in fp64 (1/128 folded in)
//  kernel 1 (stage A, f32 WMMA 16x16x4): t[n,r][b,u*16+v] = sum_w W_n*x
//            - weights staged to LDS via TDM tensor_load_to_lds (pad feature
//              produces the bank-conflict-free 272-dword row stride), async
//              overlapped with the x-tile transpose staging.
//            - batch chunked by 256 so t (52.5MB) stays resident in 192MB L2.
//  kernel 2 (stage B): apply the tiny w3j contraction; float4-coalesced
//            512MB output write (the roofline floor: ~22us @ 23.3 TB/s).
// ---------------------------------------------------------------------------

typedef __attribute__((ext_vector_type(2))) float v2f;
typedef __attribute__((ext_vector_type(8))) float v8f;
typedef __attribute__((ext_vector_type(4))) unsigned int u32x4;
typedef __attribute__((ext_vector_type(4))) int i32x4;
typedef __attribute__((ext_vector_type(8))) int i32x8;

#define CHUNK_B   256
#define TPLANE    ((size_t)(CHUNK_B) * 256)   // floats per (instr,r) plane = 65536
#define W_STRIDE  272                          // padded LDS stride for weights
#define W_LDSF    (128 * W_STRIDE)             // 34816 floats
#define X_STRIDE  132                          // padded LDS stride for x rows

// ---- instruction tables (order: i outer, then j, then k; triangle rule) ----
__constant__ int C_I[40] = {0,0,0,0, 1,1,1,1,1,1,1,1,1, 2,2,2,2,2,2,2,2,2,2,2,
                            3,3,3,3,3,3,3,3,3,3, 4,4,4,4,4,4};
__constant__ int C_J[40] = {0,1,2,3, 0,1,1,1,2,2,2,3,3, 0,1,1,1,2,2,2,2,3,3,3,
                            0,1,1,2,2,2,3,3,3,3, 1,2,2,3,3,3};
__constant__ int C_K[40] = {0,1,2,3, 1,0,1,2,1,2,3,2,3, 2,1,2,3,0,1,2,3,1,2,3,
                            3,2,3,1,2,3,0,1,2,3, 3,2,3,1,2,3};
// prefix sums of (2i+1): r-plane base per instruction (total 210 planes)
__constant__ int C_RBASE[40] = {0,1,2,3, 4,7,10,13,16,19,22,25,28,
                                31,36,41,46,51,56,61,66,71,76,81,
                                86,93,100,107,114,121,128,135,142,149,
                                156,165,174,183,192,201};
// per (j,k) block: contributing instruction lists
__constant__ int C_JK_CNT[16]  = {1,1,1,1, 1,3,3,3, 1,3,5,4, 1,3,4,5};
__constant__ int C_JK_LIST[80] = {
  0,0,0,0,0,   4,0,0,0,0,   13,0,0,0,0,  24,0,0,0,0,
  5,0,0,0,0,   1,6,14,0,0,  7,15,25,0,0, 16,26,34,0,0,
  17,0,0,0,0,  8,18,27,0,0, 2,9,19,28,35,10,20,29,36,0,
  30,0,0,0,0,  21,31,37,0,0,11,22,32,38,0,3,12,23,33,39};

constexpr int XOFFS[5] = {0, 128, 512, 1152, 2048};  // float offset of l-block in a row
constexpr int IBASE[5] = {0, 4, 13, 24, 34};         // first instr index per i

// ===========================================================================
// Kernel 0: e3nn real Wigner-3j tensors in fp64 (scaled by 1/(128*||C||_F))
// ===========================================================================
__device__ inline double2 cmul(double2 a, double2 b) {
  return make_double2(a.x * b.x - a.y * b.y, a.x * b.y + a.y * b.x);
}

__device__ double su2_cg(int j1, int m1, int j2, int m2, int j3, int m3) {
  if (m3 != m1 + m2) return 0.0;
  const double F[13] = {1., 1., 2., 6., 24., 120., 720., 5040., 40320.,
                        362880., 3628800., 39916800., 479001600.};
  int vmin = -j1 + j2 + m3; if (-j1 + m1 > vmin) vmin = -j1 + m1; if (vmin < 0) vmin = 0;
  int vmax = j2 + j3 + m1; if (j3 - j1 + j2 < vmax) vmax = j3 - j1 + j2;
  if (j3 + m3 < vmax) vmax = j3 + m3;
  double C = sqrt((2.0 * j3 + 1.0) * F[j3 + j1 - j2] * F[j3 - j1 + j2] * F[j1 + j2 - j3] *
                  F[j3 + m3] * F[j3 - m3] /
                  (F[j1 + j2 + j3 + 1] * F[j1 - m1] * F[j1 + m1] * F[j2 - m2] * F[j2 + m2]));
  double S = 0.0;
  for (int v = vmin; v <= vmax; ++v) {
    double term = F[j2 + j3 + m1 - v] * F[j1 - m1 + v] /
                  (F[v] * F[j3 - j1 + j2 - v] * F[j3 + m3 - v] * F[v + j1 - j2 - m3]);
    S += ((v + j2 + m2) & 1) ? -term : term;
  }
  return C * S;
}

// q_real_to_complex(l)[row][col], including the (-i)^l phase
__device__ double2 q_entry(int l, int row, int col) {
  const double s = 0.70710678118654752440;
  double re = 0.0, im = 0.0;
  int m = row - l;
  if (m < 0) {
    if (col == 2 * l - row) re = s;
    else if (col == row)    im = -s;
  } else if (m == 0) {
    if (col == l) re = 1.0;
  } else {
    double sg = (m & 1) ? -1.0 : 1.0;
    if (col == row)              re = sg * s;
    else if (col == 2 * l - row) im = sg * s;
  }
  double2 v = make_double2(re, im);
  switch (l & 3) {               // multiply by (-i)^l
    case 1: v = make_double2(v.y, -v.x); break;
    case 2: v = make_double2(-v.x, -v.y); break;
    case 3: v = make_double2(-v.y, v.x); break;
    default: break;
  }
  return v;
}

__global__ void w3j_init(float* __restrict__ w3j) {
  __shared__ double vals[448];
  __shared__ double scale_sh;
  int n = blockIdx.x;
  int l1 = C_J[n], l2 = C_K[n], l3 = C_I[n];  // wigner_3j(j, k, i)
  int d1 = 2 * l1 + 1, d2 = 2 * l2 + 1, d3 = 2 * l3 + 1;
  int tot = d1 * d2 * d3;
  for (int e = threadIdx.x; e < tot; e += blockDim.x) {
    int a = e / (d2 * d3); int rem = e - a * d2 * d3;
    int bq = rem / d3;     int cr = rem - bq * d3;
    double re = 0.0;
    for (int ii = 0; ii < d1; ++ii) {
      double2 q1 = q_entry(l1, ii, a);
      if (q1.x == 0.0 && q1.y == 0.0) continue;
      for (int kx = 0; kx < d2; ++kx) {
        double2 q2 = q_entry(l2, kx, bq);
        if (q2.x == 0.0 && q2.y == 0.0) continue;
        double2 q12 = cmul(q1, q2);
        for (int nn = 0; nn < d3; ++nn) {
          double cg = su2_cg(l1, ii - l1, l2, kx - l2, l3, nn - l3);
          if (cg == 0.0) continue;
          double2 q3 = q_entry(l3, nn, cr);
          q3.y = -q3.y;                       // conj(Q3[n][c])
          double2 p = cmul(q12, q3);
          re += p.x * cg;                     // real part only survives
        }
      }
    }
    vals[e] = re;
  }
  __syncthreads();
  if (threadIdx.x == 0) {
    double s = 0.0;
    for (int e = 0; e < tot; ++e) s += vals[e] * vals[e];
    scale_sh = (1.0 / 128.0) / sqrt(s);       // fold the /MUL_IN here
  }
  __syncthreads();
  for (int e = threadIdx.x; e < tot; e += blockDim.x)
    w3j[n * 512 + e] = (float)(vals[e] * scale_sh);
}

// ===========================================================================
// Kernel 1 (stage A): t[(rbase_n+r)*65536 + b*256 + u*16+v] =
//                     sum_w W_n[w,u,v] * x_i[b,w,r]
// grid = (16 b-tiles, #instrs of this i), block = 512 (16 waves, wave = one u)
// Weights (128x256, contiguous) staged to LDS by the Tensor Data Mover with
// LDS padding 16 dwords every 256 dwords -> row stride 272 (bank-conflict free)
// ===========================================================================
#define HAS_TDM __has_builtin(__builtin_amdgcn_tensor_load_to_lds)

template <int DI>
__global__ void __launch_bounds__(512)
stageA(const float* __restrict__ x, const float* __restrict__ wflat,
       float* __restrict__ t) {
  constexpr int I  = (DI - 1) / 2;
  constexpr int XO = XOFFS[I];
  constexpr int LEN = 128 * DI;
  extern __shared__ float smem[];
  float* w_lds = smem;                 // 128 x 256 weights, padded stride 272
  float* x_lds = smem + W_LDSF;        // [r][b][w] tile, padded stride 132

  const int n     = IBASE[I] + blockIdx.y;
  const int btile = blockIdx.x * 16;
  const int tid   = threadIdx.x;
  const int lane  = tid & 31, u = tid >> 5;   // wave index == u

  const float* wsrc = wflat + (size_t)n * 32768;

#if HAS_TDM
  // ---- async TDM load of the 256x128 weight tile, issued by wave 0 only ----
  if (u == 0) {
    uint64_t ga = (uint64_t)(uintptr_t)wsrc;
    uint32_t la = (uint32_t)(uintptr_t)w_lds;   // low 32 bits = LDS byte offset
    u32x4 g0 = { 1u,                                           // count=1
                 la,                                           // lds_addr
                 (unsigned)(ga & 0xffffffffu),                 // global_addr lo
                 (unsigned)((ga >> 32) & 0x01ffffffu) | (2u << 30) }; // hi|type=2
    // g1: data_size=4B (2<<16), pad_enable (1<<20), pad_interval=7 (256 dw),
    //     pad_amount=15 (16 dw); tensor_dim0=256, tensor_dim1=128,
    //     tile_dim0=256, tile_dim1=128, tensor_dim0_stride=256
    i32x8 g1 = { (int)0x1FD20000,
                 (int)0x01000000,   // tensor_dim0=256 -> bits[63:48]
                 (int)0x00800000,   // tensor_dim1=128 -> bits[95:80]
                 (int)0x01000000,   // tile_dim0=256   -> bits[127:112]
                 (int)0x00000080,   // tile_dim1=128   -> bits[143:128]
                 (int)0x00000100,   // tensor_dim0_stride=256 -> bits[191:160]
                 0, 0 };
    i32x4 gz4 = {0, 0, 0, 0};
#if defined(__clang_major__) && __clang_major__ >= 23
    i32x8 gz8 = {0, 0, 0, 0, 0, 0, 0, 0};
    __builtin_amdgcn_tensor_load_to_lds(g0, g1, gz4, gz4, gz8, 0);
#else
    __builtin_amdgcn_tensor_load_to_lds(g0, g1, gz4, gz4, 0);
#endif
  }
#else
  // ---- fallback: cooperative staged load with padded stride ----
  for (int s = tid; s < 32768; s += 512) {
    int wr = s >> 8, col = s & 255;
    w_lds[wr * W_STRIDE + col] = wsrc[s];
  }
#endif

  // stage x tile [16 b][128 w][DI r] -> x_lds[(r*16+b)*132 + w]  (float4 loads)
  constexpr int LEN4 = LEN / 4;
  for (int s4 = tid; s4 < 16 * LEN4; s4 += 512) {
    int b = s4 / LEN4; int i4 = (s4 - b * LEN4) * 4;
    float4 val = *reinterpret_cast<const float4*>(
        &x[(size_t)(btile + b) * 3200 + XO + i4]);
    float tmp[4] = {val.x, val.y, val.z, val.w};
#pragma unroll
    for (int e = 0; e < 4; ++e) {
      int idx = i4 + e; int ww = idx / DI; int r = idx - ww * DI;
      x_lds[(r * 16 + b) * X_STRIDE + ww] = tmp[e];
    }
  }

#if HAS_TDM
  if (u == 0) {
#if __has_builtin(__builtin_amdgcn_s_wait_tensorcnt)
    __builtin_amdgcn_s_wait_tensorcnt(0);
#else
    asm volatile("s_wait_tensorcnt 0x0" ::: "memory");
#endif
  }
#endif
  __syncthreads();

  const int mrow = lane & 15, khalf = lane >> 4;   // A/B lane mapping for 16x16x4

  v8f acc[DI];
  const v8f zero = {0.f, 0.f, 0.f, 0.f, 0.f, 0.f, 0.f, 0.f};
#pragma unroll
  for (int r = 0; r < DI; ++r) acc[r] = zero;

#pragma unroll 4
  for (int kk = 0; kk < 32; ++kk) {                // K = 128 = 32 x 4
    const int wrow = kk * 4 + khalf * 2;
    v2f bfrag;                                     // B[w, u*16+v], reused over r
    bfrag.x = w_lds[wrow * W_STRIDE + u * 16 + mrow];
    bfrag.y = w_lds[(wrow + 1) * W_STRIDE + u * 16 + mrow];
#pragma unroll
    for (int r = 0; r < DI; ++r) {
      v2f afrag = *(const v2f*)&x_lds[(r * 16 + mrow) * X_STRIDE + wrow];
      acc[r] = __builtin_amdgcn_wmma_f32_16x16x4_f32(
          false, afrag, false, bfrag, (short)0, acc[r], false, false);
    }
  }

  const int rbase = C_RBASE[n];
#pragma unroll
  for (int r = 0; r < DI; ++r) {
    float* dst = t + (size_t)(rbase + r) * TPLANE;
#pragma unroll
    for (int g = 0; g < 8; ++g) {                  // D: vgpr g = rows g / g+8
      dst[(size_t)(btile + g + khalf * 8) * 256 + u * 16 + mrow] = acc[r][g];
    }
  }
}

// ===========================================================================
// Kernel 2 (stage B): apply w3j, write output rows with float4 stores.
// block = 256 threads = 4 rows x 64 col-groups; grid.x = CHUNK_B*256/4
// ===========================================================================
__global__ void __launch_bounds__(256)
stageB(const float* __restrict__ t, const float* __restrict__ w3j,
       float* __restrict__ out) {
  const int row = blockIdx.x * 4 + (threadIdx.x >> 6);
  const int c0  = (threadIdx.x & 63) * 4;

  const int bL = row >> 8, J = row & 255;
  const int j  = (J < 16) ? 0 : ((J < 64) ? 1 : ((J < 144) ? 2 : 3));
  const int dj = 2 * j + 1;
  const int lj = J - 16 * j * j;
  const int u = lj / dj, p = lj - u * dj;
  const size_t tbase = (size_t)bL * 256 + u * 16;

  float acc[4];
#pragma unroll
  for (int e = 0; e < 4; ++e) {
    const int c  = c0 + e;
    const int k  = (c < 16) ? 0 : ((c < 64) ? 1 : ((c < 144) ? 2 : 3));
    const int dk = 2 * k + 1;
    const int lk = c - 16 * k * k;
    const int v = lk / dk, q = lk - v * dk;
    const int jk = j * 4 + k;
    const int cnt = C_JK_CNT[jk];
    float a = 0.f;
    for (int s = 0; s < cnt; ++s) {
      const int n  = C_JK_LIST[jk * 5 + s];
      const int di = 2 * C_I[n] + 1;
      const float* w3 = w3j + n * 512 + (p * dk + q) * di;
      const float* tp = t + (size_t)C_RBASE[n] * TPLANE + tbase + v;
      for (int r = 0; r < di; ++r) a += tp[(size_t)r * TPLANE] * w3[r];
    }
    acc[e] = a;
  }
  *reinterpret_cast<float4*>(&out[(size_t)row * 256 + c0]) =
      make_float4(acc[0], acc[1], acc[2], acc[3]);
}

// ===========================================================================
static inline size_t smemBytes(int DI) {
  return (size_t)(W_LDSF + DI * 16 * X_STRIDE) * sizeof(float);  // max ~215 KB
}

extern "C" void kernel_launch(void* const* d_in, const int* in_sizes, int n_in,
                              void* d_out, int out_size, void* d_ws, size_t ws_size,
                              hipStream_t stream) {
  (void)in_sizes; (void)n_in; (void)out_size; (void)ws_size;
  const float* x = (const float*)d_in[0];   // [2048, 3200]
  const float* w = (const float*)d_in[1];   // path weights (+ unused biases)
  float* out = (float*)d_out;               // [2048, 256, 256]
  float* w3j = (float*)d_ws;                // 40 x 512 floats
  float* t   = (float*)d_ws + 32768;        // 210 planes x 65536 floats (~52.5MB)

  w3j_init<<<40, 256, 0, stream>>>(w3j);

  for (int ch = 0; ch < 2048 / CHUNK_B; ++ch) {
    const float* xc = x + (size_t)ch * CHUNK_B * 3200;
    float* oc = out + (size_t)ch * CHUNK_B * 65536;
    stageA<1><<<dim3(CHUNK_B / 16, 4),  512, smemBytes(1), stream>>>(xc, w, t);
    stageA<3><<<dim3(CHUNK_B / 16, 9),  512, smemBytes(3), stream>>>(xc, w, t);
    stageA<5><<<dim3(CHUNK_B / 16, 11), 512, smemBytes(5), stream>>>(xc, w, t);
    stageA<7><<<dim3(CHUNK_B / 16, 10), 512, smemBytes(7), stream>>>(xc, w, t);
    stageA<9><<<dim3(CHUNK_B / 16, 6),  512, smemBytes(9), stream>>>(xc, w, t);
    stageB<<<CHUNK_B * 256 / 4, 256, 0, stream>>>(t, w3j, oc);
  }
}